// Encoder_84112639525135
// MI455X (gfx1250) — compile-verified
//
#include <hip/hip_runtime.h>
#include <hip/hip_bf16.h>
#include <cstdint>
#include <cstddef>

// ---------------- dims ----------------
#define NQ    2500
#define Dm    256
#define NHh   8
#define DHd   32
#define CAMS  6
#define Zz    4
#define SLl   4
#define SPp   2
#define NSs   2
#define TPp   4
#define SKV   14960

typedef __attribute__((ext_vector_type(2))) float v2f;
typedef __attribute__((ext_vector_type(8))) float v8f;

// ---------------- generic fp32 WMMA GEMM:  C[M,N] = A[M,K] * W[K,N] + bias ----------------
// One wave computes a 16x64 tile: 4 independent V_WMMA_F32_16X16X4_F32 accumulators per
// k-step (breaks the accumulator RAW chain, 4x reuse of the A fragment).
// A fragment (16x4): lanes 0-15 hold {K0,K1} of row M=lane; lanes 16-31 hold {K2,K3}.
// B fragment (4x16): lanes 0-15 hold {B[K0][n],B[K1][n]}; lanes 16-31 hold {B[K2][n],B[K3][n]}, n=lane&15.
__global__ void gemm_bias_wmma_kernel(const float* __restrict__ A, const float* __restrict__ W,
                                      const float* __restrict__ bias, float* __restrict__ C,
                                      int M, int K, int N) {
  int wave = blockIdx.x * (blockDim.x >> 5) + (threadIdx.x >> 5);
  int lane = threadIdx.x & 31;
  int tilesN = N >> 6;                           // 64-wide output tiles (all N here are multiples of 64)
  int tilesM = (M + 15) >> 4;
  if (wave >= tilesM * tilesN) return;           // uniform per wave -> EXEC stays all-ones
  int tm = (wave / tilesN) << 4;
  int tn = (wave % tilesN) << 6;

  int row   = lane & 15;
  int khalf = (lane >> 4) << 1;                  // 0 or 2
  int m = tm + row; if (m >= M) m = M - 1;       // clamped load row (stores are guarded)
  int n0 = tn + row;

  v8f acc[4];
#pragma unroll
  for (int t = 0; t < 4; ++t) acc[t] = (v8f){0.f,0.f,0.f,0.f,0.f,0.f,0.f,0.f};

  for (int k0 = 0; k0 < K; k0 += 4) {
    int ka = k0 + khalf;
    v2f a; a.x = A[(size_t)m * K + ka]; a.y = A[(size_t)m * K + ka + 1];
    const float* w0 = W + (size_t)ka * N + n0;
    const float* w1 = W + (size_t)(ka + 1) * N + n0;
    v2f b0; b0.x = w0[0];  b0.y = w1[0];
    v2f b1; b1.x = w0[16]; b1.y = w1[16];
    v2f b2; b2.x = w0[32]; b2.y = w1[32];
    v2f b3; b3.x = w0[48]; b3.y = w1[48];
    acc[0] = __builtin_amdgcn_wmma_f32_16x16x4_f32(false, a, false, b0, (short)0, acc[0], false, false);
    acc[1] = __builtin_amdgcn_wmma_f32_16x16x4_f32(false, a, false, b1, (short)0, acc[1], false, false);
    acc[2] = __builtin_amdgcn_wmma_f32_16x16x4_f32(false, a, false, b2, (short)0, acc[2], false, false);
    acc[3] = __builtin_amdgcn_wmma_f32_16x16x4_f32(false, a, false, b3, (short)0, acc[3], false, false);
  }

  int mbase = tm + ((lane >> 4) << 3);           // lanes 16-31 hold rows r+8
#pragma unroll
  for (int t = 0; t < 4; ++t) {
    int nOut = tn + t * 16 + (lane & 15);
    float bv = bias ? bias[nOut] : 0.f;
#pragma unroll
    for (int r = 0; r < 8; ++r) {
      int mo = mbase + r;
      if (mo < M) C[(size_t)mo * N + nOut] = acc[t][r] + bv;
    }
  }
}

// ---------------- concat [temp_query | key_hist0 | key_hist1] -> [NQ,768] ----------------
__global__ void build_qcat_kernel(const float* __restrict__ tq, const float* __restrict__ kh,
                                  float* __restrict__ qcat) {
  int idx = blockIdx.x * blockDim.x + threadIdx.x;
  if (idx >= NQ * 768) return;
  int q = idx / 768, col = idx - q * 768;
  float v;
  if (col < 256)      v = tq[q * 256 + col];
  else if (col < 512) v = kh[q * 256 + (col - 256)];
  else                v = kh[NQ * 256 + q * 256 + (col - 512)];
  qcat[idx] = v;
}

// ---------------- softmax over contiguous groups ----------------
__global__ void softmax_groups_kernel(const float* __restrict__ in, float* __restrict__ out,
                                      int ngroups, int gsize) {
  int g = blockIdx.x * blockDim.x + threadIdx.x;
  if (g >= ngroups) return;
  const float* p = in + (size_t)g * gsize;
  float* q = out + (size_t)g * gsize;
  float mx = p[0];
  for (int i = 1; i < gsize; ++i) mx = fmaxf(mx, p[i]);
  float s = 0.f;
  for (int i = 0; i < gsize; ++i) { float e = expf(p[i] - mx); q[i] = e; s += e; }
  float inv = 1.f / s;
  for (int i = 0; i < gsize; ++i) q[i] *= inv;
}

// ---------------- bilinear sample (zero padding, align_corners=False convention) ----------------
__device__ __forceinline__ float bilin_sample(const float* __restrict__ img, int H, int W,
                                              int ch, float x, float y) {
  float x0f = floorf(x), y0f = floorf(y);
  int x0 = (int)x0f, y0 = (int)y0f;
  float wx = x - x0f, wy = y - y0f;
  float v00 = 0.f, v10 = 0.f, v01 = 0.f, v11 = 0.f;
  bool xa = (x0 >= 0) & (x0 < W), xb = (x0 + 1 >= 0) & (x0 + 1 < W);
  bool ya = (y0 >= 0) & (y0 < H), yb = (y0 + 1 >= 0) & (y0 + 1 < H);
  if (xa & ya) v00 = img[((size_t)(y0 * W + x0)) * Dm + ch];
  if (xb & ya) v10 = img[((size_t)(y0 * W + x0 + 1)) * Dm + ch];
  if (xa & yb) v01 = img[((size_t)((y0 + 1) * W + x0)) * Dm + ch];
  if (xb & yb) v11 = img[((size_t)((y0 + 1) * W + x0 + 1)) * Dm + ch];
  return v00 * (1.f - wx) * (1.f - wy) + v10 * wx * (1.f - wy)
       + v01 * (1.f - wx) * wy        + v11 * wx * wy;
}

// ---------------- temporal deformable sampling + mean over NS ----------------
// grid: NQ blocks, 256 threads (h = tid/32, d = tid&31)
__global__ void temporal_msda_kernel(const float* __restrict__ tval,  // [2*NQ, 256]
                                     const float* __restrict__ toff,  // [NQ, 128]
                                     const float* __restrict__ taw,   // [NQ, 64] softmaxed
                                     const float* __restrict__ tref,  // [NQ, 1, 2]
                                     float* __restrict__ outp) {      // [NQ, 256]
  int q = blockIdx.x;
  int h = threadIdx.x >> 5, d = threadIdx.x & 31;
  int ch = h * 32 + d;
  float rx = tref[q * 2 + 0], ry = tref[q * 2 + 1];
  float acc = 0.f;
#pragma unroll
  for (int s = 0; s < NSs; ++s) {
    const float* img = tval + (size_t)s * NQ * Dm;
#pragma unroll
    for (int p = 0; p < TPp; ++p) {
      float ox = toff[q * 128 + s * 64 + h * 8 + p * 2 + 0] * (1.f / 50.f);
      float oy = toff[q * 128 + s * 64 + h * 8 + p * 2 + 1] * (1.f / 50.f);
      float x = (rx + ox) * 50.f - 0.5f;
      float y = (ry + oy) * 50.f - 0.5f;
      acc += taw[q * 64 + s * 32 + h * 4 + p] * bilin_sample(img, 50, 50, ch, x, y);
    }
  }
  outp[q * Dm + ch] = acc * 0.5f;   // mean over NS=2
}

// ---------------- spatial deformable sampling per camera ----------------
// grid: (NQ, CAMS), 256 threads (h = tid/32, d = tid&31)
__global__ void spatial_msda_kernel(const float* __restrict__ sval,  // [CAMS*SKV, 256]
                                    const float* __restrict__ soff,  // [NQ, 512]
                                    const float* __restrict__ saw,   // [NQ, 256] softmaxed
                                    const float* __restrict__ sref,  // [CAMS, NQ, Z, 2]
                                    float* __restrict__ outp) {      // [CAMS, NQ, 256]
  int q = blockIdx.x, c = blockIdx.y;
  int h = threadIdx.x >> 5, d = threadIdx.x & 31;
  int ch = h * 32 + d;
  const int LH[4] = {64, 32, 16, 8};
  const int LW[4] = {176, 88, 44, 22};
  const int LS[4] = {0, 11264, 14080, 14784};
  const float* valc = sval + (size_t)c * SKV * Dm;
  const float* refq = sref + ((size_t)c * NQ + q) * Zz * 2;
  const float* offq = soff + (size_t)q * 512 + h * 64;
  const float* awq  = saw  + (size_t)q * 256 + h * 32;
  float acc = 0.f;
  for (int l = 0; l < SLl; ++l) {
    int H = LH[l], W = LW[l];
    float invW = 1.f / (float)W, invH = 1.f / (float)H;
    const float* img = valc + (size_t)LS[l] * Dm;
#pragma unroll
    for (int p = 0; p < SPp; ++p)
#pragma unroll
      for (int z = 0; z < Zz; ++z) {
        float ox = offq[l * 16 + p * 8 + z * 2 + 0] * invW;
        float oy = offq[l * 16 + p * 8 + z * 2 + 1] * invH;
        float x = (refq[z * 2 + 0] + ox) * (float)W - 0.5f;
        float y = (refq[z * 2 + 1] + oy) * (float)H - 0.5f;
        acc += awq[l * 8 + p * 4 + z] * bilin_sample(img, H, W, ch, x, y);
      }
  }
  outp[((size_t)c * NQ + q) * Dm + ch] = acc;
}

// ---------------- masked mean across cameras ----------------
__global__ void combine_cams_kernel(const float* __restrict__ smsda,        // [CAMS, NQ, 256]
                                    const unsigned char* __restrict__ mask, // [CAMS, NQ, 4] bool
                                    float* __restrict__ slots) {            // [NQ, 256]
  int idx = blockIdx.x * blockDim.x + threadIdx.x;
  if (idx >= NQ * Dm) return;
  int q = idx >> 8;
  int chn = idx & 255;
  float sum = 0.f, msum = 0.f;
#pragma unroll
  for (int c = 0; c < CAMS; ++c) {
    const unsigned char* mk = mask + ((size_t)c * NQ + q) * 4;
    float m = (mk[0] | mk[1] | mk[2] | mk[3]) ? 1.f : 0.f;
    msum += m;
    sum += m * smsda[((size_t)c * NQ + q) * Dm + chn];
  }
  slots[idx] = sum / fmaxf(msum, 1.f);
}

// ---------------- fused residual + LayerNorm, D=256, one block per row ----------------
__global__ void add_ln_kernel(const float* __restrict__ A, const float* __restrict__ B,
                              const float* __restrict__ g, const float* __restrict__ b,
                              float* __restrict__ out) {
  int row = blockIdx.x, t = threadIdx.x;
  float v = A[(size_t)row * Dm + t] + B[(size_t)row * Dm + t];
  __shared__ float red[Dm];
  red[t] = v; __syncthreads();
  for (int s = 128; s > 0; s >>= 1) { if (t < s) red[t] += red[t + s]; __syncthreads(); }
  float mean = red[0] * (1.f / Dm);
  __syncthreads();
  float dv = v - mean;
  red[t] = dv * dv; __syncthreads();
  for (int s = 128; s > 0; s >>= 1) { if (t < s) red[t] += red[t + s]; __syncthreads(); }
  float var = red[0] * (1.f / Dm);
  out[(size_t)row * Dm + t] = dv * rsqrtf(var + 1e-5f) * g[t] + b[t];
}

// ---------------- host orchestration ----------------
static inline void launch_gemm(const float* A, const float* W, const float* b, float* C,
                               int M, int K, int N, hipStream_t stream) {
  int tiles = ((M + 15) / 16) * (N / 64);
  int blocks = (tiles + 3) / 4;
  gemm_bias_wmma_kernel<<<blocks, 128, 0, stream>>>(A, W, b, C, M, K, N);
}

extern "C" void kernel_launch(void* const* d_in, const int* in_sizes, int n_in,
                              void* d_out, int out_size, void* d_ws, size_t ws_size,
                              hipStream_t stream) {
  const float* temp_query = (const float*)d_in[0];
  const float* key_hist   = (const float*)d_in[1];
  const float* value_hist = (const float*)d_in[2];
  const float* embed_feat = (const float*)d_in[3];
  const float* sref_cam   = (const float*)d_in[4];
  const float* tref       = (const float*)d_in[5];
  const float* t_off_W = (const float*)d_in[6];  const float* t_off_b = (const float*)d_in[7];
  const float* t_aw_W  = (const float*)d_in[8];  const float* t_aw_b  = (const float*)d_in[9];
  const float* t_val_W = (const float*)d_in[10]; const float* t_val_b = (const float*)d_in[11];
  const float* t_out_W = (const float*)d_in[12]; const float* t_out_b = (const float*)d_in[13];
  const float* s_off_W = (const float*)d_in[14]; const float* s_off_b = (const float*)d_in[15];
  const float* s_aw_W  = (const float*)d_in[16]; const float* s_aw_b  = (const float*)d_in[17];
  const float* s_val_W = (const float*)d_in[18]; const float* s_val_b = (const float*)d_in[19];
  const float* s_out_W = (const float*)d_in[20]; const float* s_out_b = (const float*)d_in[21];
  const float* ffn_W   = (const float*)d_in[22]; const float* ffn_b   = (const float*)d_in[23];
  const float* ln1_g = (const float*)d_in[24]; const float* ln1_b = (const float*)d_in[25];
  const float* ln2_g = (const float*)d_in[26]; const float* ln2_b = (const float*)d_in[27];
  const float* ln3_g = (const float*)d_in[28]; const float* ln3_b = (const float*)d_in[29];
  const unsigned char* bev_mask = (const unsigned char*)d_in[30];
  float* out = (float*)d_out;

  // ---- workspace layout (persistent region + reusable scratch region) ----
  char* base = (char*)d_ws;
  size_t o = 0;
  auto alloc = [&](size_t bytes) { size_t r = o; o += (bytes + 255) & ~(size_t)255; return r; };
  const size_t V = NQ * Dm * sizeof(float);          // 2.56 MB
  size_t o_x1    = alloc(V);
  size_t o_soff  = alloc((size_t)NQ * 512 * 4);
  size_t o_sawl  = alloc(V);
  size_t o_saw   = alloc(V);
  size_t o_smsda = alloc((size_t)CAMS * NQ * Dm * 4);
  size_t o_slots = alloc(V);
  size_t o_sattn = alloc(V);
  size_t o_x2    = alloc(V);
  size_t o_x3    = alloc(V);
  size_t scratch = o;                                 // reused between stages
  // temporal scratch
  size_t so = scratch;
  auto salloc = [&](size_t bytes) { size_t r = so; so += (bytes + 255) & ~(size_t)255; return r; };
  size_t o_qcat = salloc((size_t)NQ * 768 * 4);
  size_t o_toff = salloc((size_t)NQ * 128 * 4);
  size_t o_tawl = salloc((size_t)NQ * 64 * 4);
  size_t o_taw  = salloc((size_t)NQ * 64 * 4);
  size_t o_tval = salloc((size_t)NSs * NQ * Dm * 4);
  size_t o_tmsd = salloc(V);
  size_t o_tatt = salloc(V);
  // spatial scratch overlaps the temporal scratch (temporal results already folded into x1)
  size_t o_sval = scratch;                            // CAMS*SKV*256 floats = ~92 MB

  float* qcat  = (float*)(base + o_qcat);
  float* toff  = (float*)(base + o_toff);
  float* tawl  = (float*)(base + o_tawl);
  float* taw   = (float*)(base + o_taw);
  float* tval  = (float*)(base + o_tval);
  float* tmsd  = (float*)(base + o_tmsd);
  float* tatt  = (float*)(base + o_tatt);
  float* x1    = (float*)(base + o_x1);
  float* soff  = (float*)(base + o_soff);
  float* sawl  = (float*)(base + o_sawl);
  float* saw   = (float*)(base + o_saw);
  float* sval  = (float*)(base + o_sval);
  float* smsda = (float*)(base + o_smsda);
  float* slots = (float*)(base + o_slots);
  float* sattn = (float*)(base + o_sattn);
  float* x2    = (float*)(base + o_x2);
  float* x3    = (float*)(base + o_x3);

  // ---- temporal self-attention ----
  build_qcat_kernel<<<(NQ * 768 + 255) / 256, 256, 0, stream>>>(temp_query, key_hist, qcat);
  launch_gemm(qcat, t_off_W, t_off_b, toff, NQ, 768, 128, stream);
  launch_gemm(qcat, t_aw_W,  t_aw_b,  tawl, NQ, 768, 64,  stream);
  softmax_groups_kernel<<<(NQ * NSs * NHh + 255) / 256, 256, 0, stream>>>(tawl, taw, NQ * NSs * NHh, 4);
  launch_gemm(value_hist, t_val_W, t_val_b, tval, NSs * NQ, Dm, Dm, stream);
  temporal_msda_kernel<<<NQ, 256, 0, stream>>>(tval, toff, taw, tref, tmsd);
  launch_gemm(tmsd, t_out_W, t_out_b, tatt, NQ, Dm, Dm, stream);
  add_ln_kernel<<<NQ, 256, 0, stream>>>(tatt, temp_query, ln1_g, ln1_b, x1);

  // ---- spatial cross-attention ----
  launch_gemm(x1, s_off_W, s_off_b, soff, NQ, Dm, 512, stream);
  launch_gemm(x1, s_aw_W,  s_aw_b,  sawl, NQ, Dm, Dm,  stream);
  softmax_groups_kernel<<<(NQ * NHh + 255) / 256, 256, 0, stream>>>(sawl, saw, NQ * NHh, 32);
  launch_gemm(embed_feat, s_val_W, s_val_b, sval, CAMS * SKV, Dm, Dm, stream);
  {
    dim3 grid(NQ, CAMS);
    spatial_msda_kernel<<<grid, 256, 0, stream>>>(sval, soff, saw, sref_cam, smsda);
  }
  combine_cams_kernel<<<(NQ * Dm + 255) / 256, 256, 0, stream>>>(smsda, bev_mask, slots);
  launch_gemm(slots, s_out_W, s_out_b, sattn, NQ, Dm, Dm, stream);
  add_ln_kernel<<<NQ, 256, 0, stream>>>(sattn, x1, ln2_g, ln2_b, x2);

  // ---- FFN + add & norm ----
  launch_gemm(x2, ffn_W, ffn_b, x3, NQ, Dm, Dm, stream);
  add_ln_kernel<<<NQ, 256, 0, stream>>>(x3, x2, ln3_g, ln3_b, out);
}